// RewardHead_56169582297090
// MI455X (gfx1250) — compile-verified
//
#include <hip/hip_runtime.h>
#include <math.h>

// ---------------------------------------------------------------------------
// RewardHead for MI455X (gfx1250).  fp32 WMMA (v_wmma_f32_16x16x4_f32) path.
// Memory-bound problem (~6MB traffic -> ~0.26us @ 23.3TB/s); all attention
// algebra folded so per-batch work is 7 small [16,64]x[64,256]-class GEMMs
// plus a [16,256]x[256,48] score GEMM, all on the fp32 WMMA pipe.
//
// All batch-independent weight products are precomputed into d_ws in WMMA
// B-operand *pair* layout: rows k and k+1 interleaved element-wise, so each
// WMMA B operand is a single aligned global_load_b64 instead of two b32.
// ---------------------------------------------------------------------------

typedef float v2f __attribute__((ext_vector_type(2)));
typedef float v8f __attribute__((ext_vector_type(8)));

// ---- workspace layout (floats); all matrices in paired-B layout -----------
#define WS_INV   0                   // 1/sigma_max(form_W)
#define WS_ZQ    16                  // [64][256]  z_embed_W @ Wq
#define WS_LK    (WS_ZQ+16384)       // [64][256]  lat_W @ Wk
#define WS_CK    (WS_LK+16384)       // [64][256]  code_W @ Wk
#define WS_LV    (WS_CK+16384)       // [64][256]  lat_W @ Wv
#define WS_CV    (WS_LV+16384)       // [64][256]  code_W @ Wv
#define WS_BQ    (WS_CV+16384)       // [256] z_embed_b @ Wq
#define WS_BKL   (WS_BQ+256)         // [256] lat_b @ Wk
#define WS_BKC   (WS_BKL+256)        // [256] code_b @ Wk
#define WS_CKT   (WS_BKC+256)        // [256][48]  (chart_emb_all @ Wk)^T / 16
#define WS_ANCT  (WS_CKT+12288)      // [64][48]   2 * anchors^T
#define WS_ANRM  (WS_ANCT+3072)      // [64]       |anchor_c|^2 (48 used)
#define WS_CVALL (WS_ANRM+64)        // [52][256]  chart_v rows + bvl,bvc,0,0
#define WS_FT    (WS_CVALL+13312)    // [64][256]  (Wo @ form_W)^T  (unscaled)
#define WS_END   (WS_FT+16384)       // ~500KB

// paired-B address: element (k, n) of a [K][stride] matrix lives at
//   base + (k&~1)*stride + 2*n + (k&1)   (rows 2j / 2j+1 interleaved)
static __device__ __forceinline__ int bpair(int k, int stride, int n) {
  return (k & ~1) * stride + 2 * n + (k & 1);
}

static __device__ __forceinline__ v8f wmma4(v2f a, v2f b, v8f c) {
  // D = A(16x4,f32) x B(4x16,f32) + C(16x16,f32)
  return __builtin_amdgcn_wmma_f32_16x16x4_f32(false, a, false, b,
                                               (short)0, c, false, false);
}
// k must be even: one 8-byte load yields (W[k][n], W[k+1][n])
static __device__ __forceinline__ v2f ldB(const float* __restrict__ W,
                                          int k, int stride, int n) {
  return *(const v2f*)(W + k * stride + 2 * n);
}
static __device__ __forceinline__ float hsum16(float s) {
  s += __shfl_xor(s, 1, 32); s += __shfl_xor(s, 2, 32);
  s += __shfl_xor(s, 4, 32); s += __shfl_xor(s, 8, 32); return s;
}
static __device__ __forceinline__ float hmax16(float s) {
  s = fmaxf(s, __shfl_xor(s, 1, 32)); s = fmaxf(s, __shfl_xor(s, 2, 32));
  s = fmaxf(s, __shfl_xor(s, 4, 32)); s = fmaxf(s, __shfl_xor(s, 8, 32));
  return s;
}

// ---------------------------------------------------------------------------
// K1: sigma_max(form_W) via power iteration on G = W^T W (64x64).
// ---------------------------------------------------------------------------
__global__ __launch_bounds__(64) void sigma_kernel(
    const float* __restrict__ formW, float* __restrict__ ws) {
  __shared__ float G[64 * 64];
  __shared__ float vv[64];
  __shared__ float yy[64];
  int t = threadIdx.x;
  for (int j = 0; j < 64; ++j) {
    float s = 0.f;
    for (int m = 0; m < 256; ++m) s += formW[m * 64 + t] * formW[m * 64 + j];
    G[t * 64 + j] = s;
  }
  vv[t] = 1.0f;
  __syncthreads();
  float sig2 = 0.f;
  for (int it = 0; it < 64; ++it) {
    float s = 0.f;
    for (int j = 0; j < 64; ++j) s += G[t * 64 + j] * vv[j];
    yy[t] = s;
    __syncthreads();
    float n = 0.f;
    for (int j = 0; j < 64; ++j) n += yy[j] * yy[j];
    n = sqrtf(n);
    sig2 = n;                       // ||G v|| with ||v||=1  ->  sigma^2
    vv[t] = yy[t] / fmaxf(n, 1e-30f);
    __syncthreads();
  }
  if (t == 0) {
    float sigma = sqrtf(sig2);
    ws[WS_INV] = 1.0f / fmaxf(sigma, 1e-8f);
  }
}

// ---------------------------------------------------------------------------
// K2: fold all batch-independent weight products into ws (paired-B layout).
// One output element per thread; each is a <=256-length dot. ~32 MFLOP total.
// ---------------------------------------------------------------------------
__global__ __launch_bounds__(256) void precompute_mats(
    const float* __restrict__ chart_emb, const float* __restrict__ chart_anchor,
    const float* __restrict__ a_chart_emb, const float* __restrict__ a_chart_anchor,
    const float* __restrict__ zW, const float* __restrict__ zb,
    const float* __restrict__ latW, const float* __restrict__ latb,
    const float* __restrict__ codeW, const float* __restrict__ codeb,
    const float* __restrict__ Wq, const float* __restrict__ Wk,
    const float* __restrict__ Wv, const float* __restrict__ Wo,
    const float* __restrict__ formW, float* __restrict__ ws) {
  int i = blockIdx.x * blockDim.x + threadIdx.x;
  if (i < 5 * 16384) {  // five [64][256] = A(64x256) @ B(256x256), K-dim = d
    int m = i >> 14, r = i & 16383, d = r >> 8, n = r & 255;
    const float* A; const float* Bm;
    if (m == 0)      { A = zW;    Bm = Wq; }
    else if (m == 1) { A = latW;  Bm = Wk; }
    else if (m == 2) { A = codeW; Bm = Wk; }
    else if (m == 3) { A = latW;  Bm = Wv; }
    else             { A = codeW; Bm = Wv; }
    float s = 0.f;
    for (int t = 0; t < 256; ++t) s += A[d * 256 + t] * Bm[t * 256 + n];
    ws[WS_ZQ + m * 16384 + bpair(d, 256, n)] = s;
    return;
  }
  i -= 5 * 16384;
  if (i < 768) {  // bias rows: zb@Wq, latb@Wk, codeb@Wk (plain layout)
    int m = i >> 8, n = i & 255;
    const float* bv; const float* Bm;
    if (m == 0)      { bv = zb;    Bm = Wq; }
    else if (m == 1) { bv = latb;  Bm = Wk; }
    else             { bv = codeb; Bm = Wk; }
    float s = 0.f;
    for (int t = 0; t < 256; ++t) s += bv[t] * Bm[t * 256 + n];
    ws[WS_BQ + i] = s;
    return;
  }
  i -= 768;
  if (i < 12288) {  // chart_kT: K-dim = dm index n2 (0..255), col c (0..47)
    int n2 = i / 48, c = i % 48;
    const float* emb = (c < 32) ? (chart_emb + c * 256) : (a_chart_emb + (c - 32) * 256);
    float s = 0.f;
    for (int t = 0; t < 256; ++t) s += emb[t] * Wk[t * 256 + n2];
    ws[WS_CKT + bpair(n2, 48, c)] = s * 0.0625f;
    return;
  }
  i -= 12288;
  if (i < 3072) {  // anchorT2: K-dim = d (0..63), col c
    int d = i / 48, c = i % 48;
    float a = (c < 32) ? chart_anchor[c * 64 + d] : a_chart_anchor[(c - 32) * 64 + d];
    ws[WS_ANCT + bpair(d, 48, c)] = 2.0f * a;
    return;
  }
  i -= 3072;
  if (i < 64) {  // |anchor_c|^2, zero padded (plain layout)
    float s = 0.f;
    if (i < 48) {
      const float* a = (i < 32) ? (chart_anchor + i * 64) : (a_chart_anchor + (i - 32) * 64);
      for (int d = 0; d < 64; ++d) s += a[d] * a[d];
    }
    ws[WS_ANRM + i] = s;
    return;
  }
  i -= 64;
  if (i < 13312) {  // chart_v_all: K-dim = token row rr (0..51), col n
    int rr = i >> 8, n = i & 255;
    float s = 0.f;
    if (rr < 32) {
      const float* e = chart_emb + rr * 256;
      for (int t = 0; t < 256; ++t) s += e[t] * Wv[t * 256 + n];
    } else if (rr < 48) {
      const float* e = a_chart_emb + (rr - 32) * 256;
      for (int t = 0; t < 256; ++t) s += e[t] * Wv[t * 256 + n];
    } else if (rr == 48) {
      for (int t = 0; t < 256; ++t) s += latb[t] * Wv[t * 256 + n];
    } else if (rr == 49) {
      for (int t = 0; t < 256; ++t) s += codeb[t] * Wv[t * 256 + n];
    }
    ws[WS_CVALL + bpair(rr, 256, n)] = s;
    return;
  }
  i -= 13312;
  if (i < 16384) {  // Ft: K-dim = d, Ft[d][n] = sum_m Wo[n,m]*formW[m,d]
    int d = i >> 8, n = i & 255;
    float s = 0.f;
    for (int m = 0; m < 256; ++m) s += Wo[n * 256 + m] * formW[m * 64 + d];
    ws[WS_FT + bpair(d, 256, n)] = s;
    return;
  }
}

// ---------------------------------------------------------------------------
// K3: main kernel. 2 waves/block, 16 batch rows per wave, fp32 WMMA.
// ---------------------------------------------------------------------------
__global__ __launch_bounds__(64) void reward_main(
    const float* __restrict__ z,    const float* __restrict__ rw,
    const float* __restrict__ az,   const float* __restrict__ arw,
    const float* __restrict__ acz,  const float* __restrict__ ctl,
    const float* __restrict__ ecov, const float* __restrict__ formb,
    const float* __restrict__ ws,   float* __restrict__ out) {
  __shared__ float qbuf[2][256 * 16];   // q^T: [col][row] per wave
  __shared__ float attn[2][16 * 56];    // per-row attention weights (52 used)
  __shared__ float rsc[2][16 * 8];      // per-row scalars: zn,dl,dc,-,t1,t2

  const int tid = threadIdx.x;
  const int wave = tid >> 5, lane = tid & 31;
  const int lrow = lane & 15, khalf = lane >> 4;
  const int rowBase = blockIdx.x * 32 + wave * 16;
  const int myrow = rowBase + lrow;

  const float invs = ws[WS_INV];
  const float* ZQ   = ws + WS_ZQ;
  const float* LK   = ws + WS_LK;
  const float* CKm  = ws + WS_CK;
  const float* LV   = ws + WS_LV;
  const float* CVm  = ws + WS_CV;
  const float* BQ   = ws + WS_BQ;
  const float* BKL  = ws + WS_BKL;
  const float* BKC  = ws + WS_BKC;
  const float* CKT  = ws + WS_CKT;
  const float* ANCT = ws + WS_ANCT;
  const float* ANRM = ws + WS_ANRM;
  const float* CVA  = ws + WS_CVALL;
  const float* FT   = ws + WS_FT;

  // ---- Pass 0: A-tiles (WMMA A layout: lane=row, k pair per half) ---------
  v2f zA[16], azA[16], aczA[16];
#pragma unroll
  for (int kk = 0; kk < 16; ++kk) {
    int off = myrow * 64 + 4 * kk + 2 * khalf;
    zA[kk]   = *(const v2f*)(z + off);
    azA[kk]  = *(const v2f*)(az + off);
    aczA[kk] = *(const v2f*)(acz + off);
  }
  {
    float zn = 0.f, dl = 0.f, dc = 0.f;
#pragma unroll
    for (int kk = 0; kk < 16; ++kk) {
      zn += zA[kk].x * zA[kk].x + zA[kk].y * zA[kk].y;
      float dx = zA[kk].x - azA[kk].x, dy = zA[kk].y - azA[kk].y;
      dl += dx * dx + dy * dy;
      dx = zA[kk].x - aczA[kk].x; dy = zA[kk].y - aczA[kk].y;
      dc += dx * dx + dy * dy;
    }
    zn += __shfl_xor(zn, 16, 32);
    dl += __shfl_xor(dl, 16, 32);
    dc += __shfl_xor(dc, 16, 32);
    if (khalf == 0) {
      rsc[wave][lrow * 8 + 0] = zn;
      rsc[wave][lrow * 8 + 1] = dl;
      rsc[wave][lrow * 8 + 2] = dc;
    }
  }

  // ---- Pass 1: q, k_lat, k_code GEMMs + running q.k dots ------------------
  v8f paccL = {}, paccC = {};
#pragma unroll 1
  for (int nc = 0; nc < 16; ++nc) {
    const int n = 16 * nc + lrow;
    v8f qc = {}, kl = {}, kc = {};
#pragma unroll
    for (int kk = 0; kk < 16; ++kk) {
      int k = 4 * kk + 2 * khalf;
      qc = wmma4(zA[kk],   ldB(ZQ,  k, 256, n), qc);
      kl = wmma4(azA[kk],  ldB(LK,  k, 256, n), kl);
      kc = wmma4(aczA[kk], ldB(CKm, k, 256, n), kc);
    }
    const float bqv = BQ[n], bklv = BKL[n], bkcv = BKC[n];
#pragma unroll
    for (int r = 0; r < 8; ++r) { qc[r] += bqv; kl[r] += bklv; kc[r] += bkcv; }
    paccL += qc * kl;
    paccC += qc * kc;
    // stage q column-major: qbuf[col*16 + row]
    float* qp = &qbuf[wave][n * 16 + khalf * 8];
    float4 lo = { qc[0], qc[1], qc[2], qc[3] };
    float4 hi = { qc[4], qc[5], qc[6], qc[7] };
    *(float4*)qp = lo;
    *(float4*)(qp + 4) = hi;
  }
  __syncthreads();

  // ---- lat/code scores ----------------------------------------------------
  float slat[8], scod[8];
#pragma unroll
  for (int r = 0; r < 8; ++r) {
    float s = hsum16(paccL[r]);
    slat[r] = s * 0.0625f - rsc[wave][(r + 8 * khalf) * 8 + 1];
    s = hsum16(paccC[r]);
    scod[r] = s * 0.0625f - rsc[wave][(r + 8 * khalf) * 8 + 2];
  }

  // ---- chart scores: q @ chart_kT  (A from LDS, K=256) --------------------
  v8f sc0 = {}, sc1 = {}, sc2 = {};
#pragma unroll 4
  for (int kk = 0; kk < 64; ++kk) {
    int k = 4 * kk + 2 * khalf;
    v2f a;
    a.x = qbuf[wave][k * 16 + lrow];
    a.y = qbuf[wave][(k + 1) * 16 + lrow];
    sc0 = wmma4(a, ldB(CKT, k, 48, lrow), sc0);
    sc1 = wmma4(a, ldB(CKT, k, 48, lrow + 16), sc1);
    sc2 = wmma4(a, ldB(CKT, k, 48, lrow + 32), sc2);
  }
  // 2*z.anchor (ANCT pre-scaled by 2)
  v8f da0 = {}, da1 = {}, da2 = {};
#pragma unroll
  for (int kk = 0; kk < 16; ++kk) {
    int k = 4 * kk + 2 * khalf;
    da0 = wmma4(zA[kk], ldB(ANCT, k, 48, lrow), da0);
    da1 = wmma4(zA[kk], ldB(ANCT, k, 48, lrow + 16), da1);
    da2 = wmma4(zA[kk], ldB(ANCT, k, 48, lrow + 32), da2);
  }

  // ---- assemble scores, softmax over 50 tokens ----------------------------
  const float an0 = ANRM[lrow], an1 = ANRM[lrow + 16], an2 = ANRM[lrow + 32];
  v8f rw0, rw1, rw2;
  float znr[8];
#pragma unroll
  for (int r = 0; r < 8; ++r) {
    int row = rowBase + r + 8 * khalf;
    rw0[r] = rw[row * 32 + lrow];
    rw1[r] = rw[row * 32 + 16 + lrow];
    rw2[r] = arw[row * 16 + lrow];
    znr[r] = rsc[wave][(r + 8 * khalf) * 8 + 0];
  }
  v8f w0, w1, w2;
  float aL[8], aC[8];
#pragma unroll
  for (int r = 0; r < 8; ++r) {
    float s0 = rw0[r] * sc0[r] + da0[r] - znr[r] - an0;
    float s1 = rw1[r] * sc1[r] + da1[r] - znr[r] - an1;
    float s2 = rw2[r] * sc2[r] + da2[r] - znr[r] - an2;
    float m = hmax16(fmaxf(fmaxf(s0, s1), s2));
    m = fmaxf(m, fmaxf(slat[r], scod[r]));
    float e0 = expf(s0 - m), e1 = expf(s1 - m), e2 = expf(s2 - m);
    float t = hsum16(e0 + e1 + e2);
    float eL = expf(slat[r] - m), eC = expf(scod[r] - m);
    float inv = 1.0f / (t + eL + eC);
    w0[r] = e0 * inv * rw0[r];   // attn * rw folds the v-side rw factor
    w1[r] = e1 * inv * rw1[r];
    w2[r] = e2 * inv * rw2[r];
    aL[r] = eL * inv;
    aC[r] = eC * inv;
  }
#pragma unroll
  for (int r = 0; r < 8; ++r) {
    int row = r + 8 * khalf;
    attn[wave][row * 56 + lrow] = w0[r];
    attn[wave][row * 56 + 16 + lrow] = w1[r];
    attn[wave][row * 56 + 32 + lrow] = w2[r];
    if (lrow == 0) {
      attn[wave][row * 56 + 48] = aL[r];
      attn[wave][row * 56 + 49] = aC[r];
      attn[wave][row * 56 + 50] = 0.f;
      attn[wave][row * 56 + 51] = 0.f;
    }
  }
  __syncthreads();

  // ---- Pass 2: feat and output contractions -------------------------------
  const float sLat = attn[wave][lrow * 56 + 48];
  const float sCod = attn[wave][lrow * 56 + 49];
  v2f azS[16], aczS[16], cA[16], eA[16];
#pragma unroll
  for (int kk = 0; kk < 16; ++kk) {
    azS[kk]  = azA[kk] * sLat;    // fold attn_lat into A operand
    aczS[kk] = aczA[kk] * sCod;
    int off = myrow * 64 + 4 * kk + 2 * khalf;
    cA[kk] = *(const v2f*)(ctl + off);
    eA[kk] = *(const v2f*)(ecov + off);
  }
  v2f wA[13];
#pragma unroll
  for (int kk = 0; kk < 13; ++kk) {
    int k = 4 * kk + 2 * khalf;
    wA[kk].x = attn[wave][lrow * 56 + k];
    wA[kk].y = attn[wave][lrow * 56 + k + 1];
  }

  v8f t1a = {}, t2a = {};
#pragma unroll 1
  for (int nc = 0; nc < 16; ++nc) {
    const int n = 16 * nc + lrow;
    v8f f = {};
#pragma unroll
    for (int kk = 0; kk < 16; ++kk) {
      int k = 4 * kk + 2 * khalf;
      f = wmma4(azS[kk],  ldB(LV,  k, 256, n), f);
      f = wmma4(aczS[kk], ldB(CVm, k, 256, n), f);
    }
#pragma unroll
    for (int kk = 0; kk < 13; ++kk) {
      int k = 4 * kk + 2 * khalf;
      f = wmma4(wA[kk], ldB(CVA, k, 256, n), f);
    }
    v8f g1 = {}, g2 = {};
#pragma unroll
    for (int kk = 0; kk < 16; ++kk) {
      int k = 4 * kk + 2 * khalf;
      g1 = wmma4(cA[kk], ldB(FT, k, 256, n), g1);
      g2 = wmma4(eA[kk], ldB(FT, k, 256, n), g2);
    }
    t1a += f * g1;
    t2a += f * g2;
  }

  // ---- per-row extras: e.ctl, e.e, form_b.ctl, form_b.e -------------------
  float ec = 0.f, ee = 0.f, fbc = 0.f, fbe = 0.f;
#pragma unroll
  for (int kk = 0; kk < 16; ++kk) {
    int d = 4 * kk + 2 * khalf;
    float fb0 = formb[d], fb1 = formb[d + 1];
    ec  += eA[kk].x * cA[kk].x + eA[kk].y * cA[kk].y;
    ee  += eA[kk].x * eA[kk].x + eA[kk].y * eA[kk].y;
    fbc += fb0 * cA[kk].x + fb1 * cA[kk].y;
    fbe += fb0 * eA[kk].x + fb1 * eA[kk].y;
  }
  ec  += __shfl_xor(ec, 16, 32);
  ee  += __shfl_xor(ee, 16, 32);
  fbc += __shfl_xor(fbc, 16, 32);
  fbe += __shfl_xor(fbe, 16, 32);

#pragma unroll
  for (int r = 0; r < 8; ++r) {
    float s1 = hsum16(t1a[r]);
    float s2 = hsum16(t2a[r]);
    if (lrow == 0) {
      rsc[wave][(r + 8 * khalf) * 8 + 4] = s1 * invs;
      rsc[wave][(r + 8 * khalf) * 8 + 5] = s2 * invs;
    }
  }
  __syncthreads();

  if (khalf == 0) {
    float t1 = rsc[wave][lrow * 8 + 4] + fbc;   // form_raw . control
    float t2 = rsc[wave][lrow * 8 + 5] + fbe;   // form_raw . e
    float coeff = (ee > 1e-8f) ? (t2 / fmaxf(ee, 1e-8f)) : 0.f;
    out[myrow] = t1 - coeff * ec;
  }
}

// ---------------------------------------------------------------------------
extern "C" void kernel_launch(void* const* d_in, const int* in_sizes, int n_in,
                              void* d_out, int out_size, void* d_ws, size_t ws_size,
                              hipStream_t stream) {
  const float* z      = (const float*)d_in[0];
  const float* rw     = (const float*)d_in[1];
  const float* az     = (const float*)d_in[2];
  const float* arw    = (const float*)d_in[3];
  const float* acz    = (const float*)d_in[4];
  const float* ctl    = (const float*)d_in[5];
  const float* ecov   = (const float*)d_in[6];
  const float* c_emb  = (const float*)d_in[7];
  const float* c_anc  = (const float*)d_in[8];
  const float* ac_emb = (const float*)d_in[9];
  const float* ac_anc = (const float*)d_in[10];
  const float* zW     = (const float*)d_in[11];
  const float* zb     = (const float*)d_in[12];
  const float* latW   = (const float*)d_in[13];
  const float* latb   = (const float*)d_in[14];
  const float* codeW  = (const float*)d_in[15];
  const float* codeb  = (const float*)d_in[16];
  const float* Wq     = (const float*)d_in[17];
  const float* Wk     = (const float*)d_in[18];
  const float* Wv     = (const float*)d_in[19];
  const float* Wo     = (const float*)d_in[20];
  const float* formW  = (const float*)d_in[21];
  const float* formb  = (const float*)d_in[22];
  float* ws = (float*)d_ws;              // needs WS_END*4 ~ 512KB
  float* out = (float*)d_out;

  sigma_kernel<<<1, 64, 0, stream>>>(formW, ws);
  precompute_mats<<<500, 256, 0, stream>>>(c_emb, c_anc, ac_emb, ac_anc,
                                           zW, zb, latW, latb, codeW, codeb,
                                           Wq, Wk, Wv, Wo, formW, ws);
  reward_main<<<128, 64, 0, stream>>>(z, rw, az, arw, acz, ctl, ecov,
                                      formb, ws, out);
}